// HandshakingKernel_84293028151980
// MI455X (gfx1250) — compile-verified
//
#include <hip/hip_runtime.h>

typedef float v2f __attribute__((ext_vector_type(2)));
typedef float v8f __attribute__((ext_vector_type(8)));

#define HID    768
#define SEQ    512
#define BATCH  2
#define MTOT   (BATCH * SEQ)          // 1024 rows of X
#define NTOT   (2 * HID)              // 1536 cols of Y (A half | C half)
#define NPAIRS ((SEQ * (SEQ + 1)) / 2)  // 131328
#define JCHUNK 32
#define JCHUNKS (SEQ / JCHUNK)        // 16

__device__ __forceinline__ float hs_tanh(float x) {
#if __has_builtin(__builtin_amdgcn_tanhf)
  return __builtin_amdgcn_tanhf(x);
#else
  return tanhf(x);
#endif
}

// ---------------------------------------------------------------------------
// Kernel 1: Y[m, n] = sum_k X[m,k] * Wc[k,n]
//   Wc[k,n] = W[k, n]        for n <  768   (left/"A" half)
//   Wc[k,n] = W[k+768, n-768] for n >= 768  (right/"C" half)
// One wave -> 16(M) x 64(N) tile via 4 accumulators of V_WMMA_F32_16X16X4_F32.
// 8 waves per block, 1024/16 * 1536/64 = 1536 wave-tiles = 192 blocks exactly.
// ---------------------------------------------------------------------------
__global__ __launch_bounds__(256) void hs_gemm(const float* __restrict__ X,
                                               const float* __restrict__ W,
                                               float* __restrict__ Y) {
  const int lane = threadIdx.x & 31;
  const int wave = threadIdx.x >> 5;
  const int tile = blockIdx.x * 8 + wave;   // 0..1535
  const int ntiles_n = NTOT / 64;           // 24
  const int mt = tile / ntiles_n;
  const int nt = tile % ntiles_n;
  const int m0 = mt * 16;
  const int n0 = nt * 64;
  const int ml = lane & 15;
  const int hi = lane >> 4;                 // 0: low half-wave, 1: high half-wave

  // A fragment (16x4 f32): lanes 0-15 rows M=0..15 with K = k0 + {0,1};
  // lanes 16-31 same rows with K = k0 + {2,3}.
  const float* aptr = X + (size_t)(m0 + ml) * HID + 2 * hi;

  // B fragment (4x16 f32): row-striped across lanes; lanes 0-15 hold rows
  // K = k0 + {0,1}, lanes 16-31 rows K = k0 + {2,3}; N = ml.
  const float* bptr[4];
#pragma unroll
  for (int s = 0; s < 4; ++s) {
    int ncol = n0 + s * 16 + ml;
    int rowoff = 0;
    if (ncol >= HID) { rowoff = HID; ncol -= HID; }
    bptr[s] = W + (size_t)(rowoff + 2 * hi) * HID + ncol;
  }

  v8f acc[4] = {};

  for (int k = 0; k < HID; k += 4) {
    v2f a;
    a.x = aptr[k];
    a.y = aptr[k + 1];
#pragma unroll
    for (int s = 0; s < 4; ++s) {
      v2f b;
      b.x = bptr[s][(size_t)k * HID];
      b.y = bptr[s][(size_t)(k + 1) * HID];
      acc[s] = __builtin_amdgcn_wmma_f32_16x16x4_f32(
          /*neg_a=*/false, a, /*neg_b=*/false, b,
          /*c_mod=*/(short)0, acc[s], /*reuse_a=*/false, /*reuse_b=*/false);
    }
  }

  // C/D layout: VGPR r holds M = r (lanes 0-15) / M = r + 8 (lanes 16-31).
#pragma unroll
  for (int s = 0; s < 4; ++s) {
    const int ncol = n0 + s * 16 + ml;
#pragma unroll
    for (int r = 0; r < 8; ++r) {
      const int m = m0 + r + 8 * hi;
      Y[(size_t)m * NTOT + ncol] = acc[s][r];
    }
  }
}

// ---------------------------------------------------------------------------
// Kernel 2: out[b, p(i,j), h] = tanh(A[b,i,h] + C[b,j,h] + bias[h])
// A = Y[., 0:768), C = Y[., 768:1536). One block = one (b, i) and a chunk of
// 32 j's; A-row + bias held in registers, output streamed with NT stores.
// HBM-write bound: 807 MB @ 23.3 TB/s ~ 35 us floor.
// ---------------------------------------------------------------------------
__global__ __launch_bounds__(256) void hs_pairs(const float* __restrict__ Y,
                                                const float* __restrict__ bias,
                                                float* __restrict__ out) {
  unsigned bid = blockIdx.x;
  const int jc = bid & (JCHUNKS - 1); bid >>= 4;   // 16 chunks
  const int i  = bid & (SEQ - 1);     bid >>= 9;   // 512 rows
  const int b  = (int)bid;                          // 2 batches

  int j0 = jc * JCHUNK;
  int j1 = j0 + JCHUNK;
  if (j1 <= i) return;          // chunk entirely below diagonal
  if (j0 < i) j0 = i;

  const int t = threadIdx.x;    // 768 = 3 * 256 elements per row

  const float* Arow = Y + ((size_t)b * SEQ + i) * NTOT;
  const float a0 = Arow[t]       + bias[t];
  const float a1 = Arow[t + 256] + bias[t + 256];
  const float a2 = Arow[t + 512] + bias[t + 512];

  // p(i,j) = i*S - i*(i-1)/2 + (j - i) = prow + j
  const long long prow =
      (long long)i * (SEQ - 1) - ((long long)i * (i - 1)) / 2;

  for (int j = j0; j < j1; ++j) {
    const float* Crow = Y + ((size_t)b * SEQ + j) * NTOT + HID;
    const float c0 = Crow[t];
    const float c1 = Crow[t + 256];
    const float c2 = Crow[t + 512];

    const size_t p = (size_t)(prow + j);
    float* o = out + ((size_t)b * NPAIRS + p) * HID;

    __builtin_nontemporal_store(hs_tanh(a0 + c0), o + t);
    __builtin_nontemporal_store(hs_tanh(a1 + c1), o + t + 256);
    __builtin_nontemporal_store(hs_tanh(a2 + c2), o + t + 512);
  }
}

extern "C" void kernel_launch(void* const* d_in, const int* in_sizes, int n_in,
                              void* d_out, int out_size, void* d_ws, size_t ws_size,
                              hipStream_t stream) {
  const float* X    = (const float*)d_in[0];  // (2, 512, 768) f32 -> (1024, 768)
  const float* W    = (const float*)d_in[1];  // (1536, 768) f32
  const float* bias = (const float*)d_in[2];  // (768,) f32
  float* out = (float*)d_out;                 // (2, 131328, 768) f32
  float* Y   = (float*)d_ws;                  // (1024, 1536) f32 = 6.3 MB

  hs_gemm<<<(MTOT / 16) * (NTOT / 64) / 8, 256, 0, stream>>>(X, W, Y);
  hs_pairs<<<BATCH * SEQ * JCHUNKS, 256, 0, stream>>>(Y, bias, out);
}